// CustomEmbedding_6734508720581
// MI455X (gfx1250) — compile-verified
//
#include <hip/hip_runtime.h>

// ---------------- toolchain probes (show up in compiler stderr) -------------
#define PSTR2(x) #x
#define PSTR(x) PSTR2(x)
#pragma message("PROBE clang=" PSTR(__clang_major__) "." PSTR(__clang_minor__))
#if __has_include(<hip/amd_detail/amd_gfx1250_TDM.h>)
#pragma message("PROBE has_TDM_header=1")
#else
#pragma message("PROBE has_TDM_header=0")
#endif

#if __has_builtin(__builtin_amdgcn_global_load_async_to_lds_b128)
#define HAVE_ASYNC_LOAD 1
#else
#define HAVE_ASYNC_LOAD 0
#endif
#if __has_builtin(__builtin_amdgcn_s_wait_asynccnt)
#define HAVE_WAIT_ASYNC 1
#else
#define HAVE_WAIT_ASYNC 0
#endif

typedef float v4f __attribute__((ext_vector_type(4)));
typedef int   v4i __attribute__((vector_size(4 * sizeof(int))));  // matches builtin proto

#define AS_GLOBAL __attribute__((address_space(1)))
#define AS_LDS    __attribute__((address_space(3)))

#define EMB_DIM   512
#define NUM_NUM   1000
#define WPB       8      // waves per 256-thread block (wave32)
#define TOK_PER_W 4      // target tokens per wave for grid sizing

template <int N>
__device__ __forceinline__ void wait_async() {
#if HAVE_WAIT_ASYNC
  __builtin_amdgcn_s_wait_asynccnt(N);
#else
  asm volatile("s_wait_asynccnt %0" ::"n"(N) : "memory");
#endif
  asm volatile("" ::: "memory");  // keep subsequent ds_loads below the wait
}

#if HAVE_ASYNC_LOAD
__device__ __forceinline__ void async_cp16(const float* gsrc, float* ldst) {
  // DMA 16B global -> LDS per lane, tracked with ASYNCcnt (no VGPR data path).
  __builtin_amdgcn_global_load_async_to_lds_b128(
      (AS_GLOBAL v4i*)gsrc, (AS_LDS v4i*)ldst, /*offset=*/0, /*cpol=*/0);
}
#else
__device__ __forceinline__ void async_cp16(const float* gsrc, float* ldst) {
  *(v4f*)ldst = *(const v4f*)gsrc;  // fallback (still correct, through VGPRs)
}
#endif

// Issue one 2KB weight row into an LDS buffer: 4 x b128 per lane (wave32).
__device__ __forceinline__ void issue_row_loads(const float* __restrict__ w,
                                                int id, float* lbase, int lane) {
  const float* src = w + (size_t)id * EMB_DIM;
#pragma unroll
  for (int j = 0; j < 4; ++j) {
    const int q = lane + 32 * j;  // float4 index within the row
    async_cp16(src + 4 * q, lbase + 4 * q);
  }
}

__global__ void __launch_bounds__(256)
embed_gather_kernel(const int* __restrict__ x, const float* __restrict__ w,
                    float* __restrict__ out, int n) {
  // double-buffered 2KB staging per wave -> 32KB/block (WGP has 320KB)
  __shared__ __align__(16) float lbuf[WPB][2][EMB_DIM];
  const int lane  = threadIdx.x & 31;
  const int wave  = threadIdx.x >> 5;
  const int gwave = blockIdx.x * WPB + wave;
  const int nwave = gridDim.x * WPB;

  int t = gwave;
  if (t >= n) return;

  // prologue: fetch first token id (wave-uniform -> SGPR) and prime pipeline
  int  id       = __builtin_amdgcn_readfirstlane(x[t]);
  bool cur_copy = (id >= NUM_NUM);
  int  b        = 0;
  if (cur_copy) issue_row_loads(w, id, &lbuf[wave][0][0], lane);

  while (t < n) {
    // ---- prefetch next token's row into the other buffer -------------------
    const int tn = t + nwave;
    int  idn      = 0;
    bool nxt_copy = false;
    if (tn < n) {
      idn      = __builtin_amdgcn_readfirstlane(x[tn]);
      nxt_copy = (idn >= NUM_NUM);
      if (nxt_copy) issue_row_loads(w, idn, &lbuf[wave][b ^ 1][0], lane);
    }

    v4f* __restrict__ dst = (v4f*)(out + (size_t)t * EMB_DIM);

    if (cur_copy) {
      // async loads complete IN ORDER: waiting to <=4 outstanding guarantees
      // the current buffer's 4 loads are done while next's stay in flight.
      if (nxt_copy) wait_async<4>(); else wait_async<0>();
      const float* lb = &lbuf[wave][b][0];
#pragma unroll
      for (int j = 0; j < 4; ++j) {
        const int q = lane + 32 * j;
        v4f v = *(const v4f*)(lb + 4 * q);             // ds_load_b128
        // output streamed once: NT stores keep the 102MB weight table in L2
        __builtin_nontemporal_store(v, dst + q);       // NT global_store_b128
      }
    } else {
      // numeric token: sin((id/1000)*(i+1)), i in [0,512) — ~2% of tokens
      const float s = (float)id * 1.0e-3f;
#pragma unroll
      for (int j = 0; j < 4; ++j) {
        const int q = lane + 32 * j;
        const float f0 = (float)(4 * q + 1);
        v4f v;
        v.x = sinf(s * f0);
        v.y = sinf(s * (f0 + 1.0f));
        v.z = sinf(s * (f0 + 2.0f));
        v.w = sinf(s * (f0 + 3.0f));
        __builtin_nontemporal_store(v, dst + q);
      }
    }

    t = tn; id = idn; cur_copy = nxt_copy; b ^= 1;
  }
}

extern "C" void kernel_launch(void* const* d_in, const int* in_sizes, int n_in,
                              void* d_out, int out_size, void* d_ws, size_t ws_size,
                              hipStream_t stream) {
  const int*   x = (const int*)d_in[0];    // (8,4096) int32 token ids
  const float* w = (const float*)d_in[1];  // (50000,512) fp32 table
  // d_in[2] (num_value) and d_in[3] (is_num) are derivable from id: ignored.
  float* out = (float*)d_out;              // (8,4096,512) fp32
  const int n = in_sizes[0];               // 32768 tokens

  int blocks = (n + WPB * TOK_PER_W - 1) / (WPB * TOK_PER_W);
  if (blocks < 1) blocks = 1;
  if (blocks > 8192) blocks = 8192;
  hipLaunchKernelGGL(embed_gather_kernel, dim3(blocks), dim3(256), 0, stream,
                     x, w, out, n);
}